// Classifier_45887430591266
// MI455X (gfx1250) — compile-verified
//
#include <hip/hip_runtime.h>

typedef __attribute__((ext_vector_type(2))) float v2f;
typedef __attribute__((ext_vector_type(8))) float v8f;

#define EMB    256
#define SENTS  128
#define BATCH  64
#define TOKS   64
#define NCLS   100
#define CPAD   112   // classes padded to 7 * 16 WMMA tiles

// ---------------------------------------------------------------------------
// Kernel 1: sents[b,s,:] = mean_l emb[tok[b,s,l], :]
// One 64-thread block per sentence; each lane owns 4 contiguous emb dims
// (float4). Token indices are block-uniform -> scalar loads; each token row
// read is 64 lanes x 16B = 1KB fully coalesced. Emb table (102MB) is
// L2-resident (192MB), so this runs at ~L2 bandwidth.
// ---------------------------------------------------------------------------
__global__ void gather_mean_kernel(const int* __restrict__ tok,
                                   const float* __restrict__ emb,
                                   float* __restrict__ sents) {
  const int bs = blockIdx.x;            // 0 .. B*S-1
  const int t  = threadIdx.x;           // 0 .. 63
  const int* tl = tok + (size_t)bs * TOKS;
  const float4* embv = (const float4*)emb;
  float4 acc = make_float4(0.f, 0.f, 0.f, 0.f);
  #pragma unroll 4
  for (int l = 0; l < TOKS; ++l) {
    const int tk = tl[l];               // uniform -> s_load
    float4 v = embv[(size_t)tk * (EMB / 4) + t];
    acc.x += v.x; acc.y += v.y; acc.z += v.z; acc.w += v.w;
  }
  const float inv = 1.0f / (float)TOKS;
  acc.x *= inv; acc.y *= inv; acc.z *= inv; acc.w *= inv;
  ((float4*)sents)[(size_t)bs * (EMB / 4) + t] = acc;
}

// ---------------------------------------------------------------------------
// Kernel 2: scores[b,c,s] = class_embs[c,:] . sents[b,s,:]   (WMMA f32 16x16x4)
// Grid (8 s-tiles, 64 batches), 224 threads = 7 waves, wave w = class tile w.
// A[m][k] = class_embs[ct*16+m][k], B[k][n] = sents[b][st*16+n][k].
// Padded classes (100..111) just recompute class 99 (clamped row pointer);
// those score rows are never read downstream, so no mask multiply is needed
// and the inner loop is pure load_b64 -> v_wmma.
// ---------------------------------------------------------------------------
__global__ void scores_wmma_kernel(const float* __restrict__ cls,
                                   const float* __restrict__ sents,
                                   float* __restrict__ scores) {
  const int st   = blockIdx.x;           // sentence tile 0..7
  const int b    = blockIdx.y;
  const int wave = threadIdx.x >> 5;     // class tile 0..6
  const int lane = threadIdx.x & 31;
  const int hl   = lane >> 4;            // half: selects K pair / D row block
  const int r15  = lane & 15;

  const int m  = wave * 16 + r15;                    // A row (class)
  const int mc = (m < NCLS) ? m : (NCLS - 1);        // clamp: memory safety only
  const int n  = st * 16 + r15;                      // B col (sentence)

  const float* arow = cls + (size_t)mc * EMB + 2 * hl;
  const float* brow = sents + ((size_t)b * SENTS + n) * EMB + 2 * hl;

  v8f acc = {};
  #pragma unroll 8
  for (int kb = 0; kb < EMB; kb += 4) {
    v2f a = *(const v2f*)(arow + kb);
    v2f bb = *(const v2f*)(brow + kb);
    acc = __builtin_amdgcn_wmma_f32_16x16x4_f32(
        /*neg_a=*/false, a, /*neg_b=*/false, bb,
        /*c_mod=*/(short)0, acc, /*reuse_a=*/false, /*reuse_b=*/false);
  }
  // D[r + 8*hl][r15] -> scores[b][wave*16 + r + 8*hl][st*16 + r15]
  float* outb = scores + (size_t)b * CPAD * SENTS;
  #pragma unroll
  for (int r = 0; r < 8; ++r) {
    const int c = wave * 16 + r + 8 * hl;
    outb[(size_t)c * SENTS + st * 16 + r15] = acc[r];
  }
}

// ---------------------------------------------------------------------------
// Kernel 3: softmax over s (128) for each (b, c<100), in place.
// ---------------------------------------------------------------------------
__global__ void softmax_kernel(float* __restrict__ scores) {
  const int c = blockIdx.x;              // 0..99
  const int b = blockIdx.y;
  const int t = threadIdx.x;             // 0..127
  float* row = scores + ((size_t)b * CPAD + c) * SENTS;
  __shared__ float red[SENTS];

  const float v = row[t];
  red[t] = v;
  __syncthreads();
  for (int off = SENTS / 2; off > 0; off >>= 1) {
    if (t < off) red[t] = fmaxf(red[t], red[t + off]);
    __syncthreads();
  }
  const float mx = red[0];
  __syncthreads();
  const float e = __expf(v - mx);
  red[t] = e;
  __syncthreads();
  for (int off = SENTS / 2; off > 0; off >>= 1) {
    if (t < off) red[t] += red[t + off];
    __syncthreads();
  }
  row[t] = e / red[0];
}

// ---------------------------------------------------------------------------
// Kernel 4: proj[b,s,c] = sents[b,s,:] . multi_weight[c,:]  (WMMA f32 16x16x4)
// (logits[b,c] = sum_s atts[b,c,s] * proj[b,s,c] + bias[c] -- algebraically
//  identical to mixes.mw, but with a smaller intermediate.)
// Padded class columns (clamped to class 99) are never read downstream.
// ---------------------------------------------------------------------------
__global__ void proj_wmma_kernel(const float* __restrict__ sents,
                                 const float* __restrict__ mw,
                                 float* __restrict__ proj) {
  const int st   = blockIdx.x;           // sentence tile 0..7 (M)
  const int b    = blockIdx.y;
  const int wave = threadIdx.x >> 5;     // class tile 0..6 (N)
  const int lane = threadIdx.x & 31;
  const int hl   = lane >> 4;
  const int r15  = lane & 15;

  const int m  = st * 16 + r15;                      // A row (sentence)
  const int n  = wave * 16 + r15;                    // B col (class)
  const int nc = (n < NCLS) ? n : (NCLS - 1);        // clamp: memory safety only

  const float* arow = sents + ((size_t)b * SENTS + m) * EMB + 2 * hl;
  const float* brow = mw + (size_t)nc * EMB + 2 * hl;

  v8f acc = {};
  #pragma unroll 8
  for (int kb = 0; kb < EMB; kb += 4) {
    v2f a = *(const v2f*)(arow + kb);
    v2f bv = *(const v2f*)(brow + kb);
    acc = __builtin_amdgcn_wmma_f32_16x16x4_f32(
        false, a, false, bv, (short)0, acc, false, false);
  }
  float* outb = proj + (size_t)b * SENTS * CPAD;
  #pragma unroll
  for (int r = 0; r < 8; ++r) {
    const int s = st * 16 + r + 8 * hl;
    outb[(size_t)s * CPAD + wave * 16 + r15] = acc[r];
  }
}

// ---------------------------------------------------------------------------
// Kernel 5: logits[b,c] = sum_s atts[b,c,s] * proj[b,s,c] + bias[c]
// ---------------------------------------------------------------------------
__global__ void logits_kernel(const float* __restrict__ atts,
                              const float* __restrict__ proj,
                              const float* __restrict__ bias,
                              float* __restrict__ out) {
  const int c = blockIdx.x;              // 0..99
  const int b = blockIdx.y;
  const int t = threadIdx.x;             // 0..127 = s
  __shared__ float red[SENTS];

  const float a = atts[((size_t)b * CPAD + c) * SENTS + t];
  const float p = proj[((size_t)b * SENTS + t) * CPAD + c];
  red[t] = a * p;
  __syncthreads();
  for (int off = SENTS / 2; off > 0; off >>= 1) {
    if (t < off) red[t] += red[t + off];
    __syncthreads();
  }
  if (t == 0) out[(size_t)b * NCLS + c] = red[0] + bias[c];
}

// ---------------------------------------------------------------------------
extern "C" void kernel_launch(void* const* d_in, const int* in_sizes, int n_in,
                              void* d_out, int out_size, void* d_ws, size_t ws_size,
                              hipStream_t stream) {
  const int*   tok  = (const int*)d_in[0];     // (64,128,64) int32
  const float* emb  = (const float*)d_in[1];   // (100000,256)
  const float* cls  = (const float*)d_in[2];   // (100,256)
  const float* mw   = (const float*)d_in[3];   // (100,256)
  const float* bias = (const float*)d_in[4];   // (100,1)
  float*       out  = (float*)d_out;           // (64,100)

  char* ws = (char*)d_ws;
  const size_t sents_bytes  = (size_t)BATCH * SENTS * EMB * sizeof(float);   // 8 MB
  const size_t scores_bytes = (size_t)BATCH * CPAD * SENTS * sizeof(float);  // 3.7 MB
  float* sents  = (float*)ws;
  float* scores = (float*)(ws + sents_bytes);
  float* proj   = (float*)(ws + sents_bytes + scores_bytes);

  gather_mean_kernel<<<BATCH * SENTS, 64, 0, stream>>>(tok, emb, sents);
  scores_wmma_kernel<<<dim3(SENTS / 16, BATCH), 224, 0, stream>>>(cls, sents, scores);
  softmax_kernel<<<dim3(NCLS, BATCH), SENTS, 0, stream>>>(scores);
  proj_wmma_kernel<<<dim3(SENTS / 16, BATCH), 224, 0, stream>>>(sents, mw, proj);
  logits_kernel<<<dim3(NCLS, BATCH), SENTS, 0, stream>>>(scores, proj, bias, out);
}